// EPR_29454885716624
// MI455X (gfx1250) — compile-verified
//
#include <hip/hip_runtime.h>

// MoE router (capacity-constrained top-k) for MI455X / gfx1250.
// Kernel 1: fp32 WMMA GEMM  x[32768,1024] * W^T[1024,4->16 padded] -> logits
// Kernel 2: per-batch softmax + 4-pass radix top-k selection (caps 512/512/1024/2048)

namespace {
constexpr int ND = 1024;  // DIM
constexpr int NE = 4;     // experts
constexpr int NB = 8;     // batch
constexpr int NT = 4096;  // tokens per batch row
}  // namespace

typedef float v2f __attribute__((ext_vector_type(2)));
typedef float v8f __attribute__((ext_vector_type(8)));

// ---------------------------------------------------------------------------
// Kernel 1: logits = x @ W^T + b  via V_WMMA_F32_16X16X4_F32.
// One wave handles a 16-token x 16-col tile (cols 0..3 = experts).
// A layout (f32 16x4): lane&15 = M row, lane>>4 picks K={0,1} vs {2,3};
// the two VGPRs of the float2 hold consecutive K. B (4x16) mirrored:
// lane&15 = N col. Lanes with n>=4 read W row 0 (valid address, finite
// garbage): D[m][n] depends only on B column n, and columns n>=4 are never
// stored, so no zeroing/predication is needed -> branch-free inner loop.
// C layout: VGPR v holds M=v (lanes 0-15) / M=v+8 (lanes 16-31), N = lane&15.
// ---------------------------------------------------------------------------
__global__ __launch_bounds__(256) void router_logits_wmma(
    const float* __restrict__ x, const float* __restrict__ W,
    const float* __restrict__ bias, float* __restrict__ logits) {
  const int flat = blockIdx.x * blockDim.x + threadIdx.x;
  const int wave = flat >> 5;
  const int lane = threadIdx.x & 31;
  const int n = lane & 15;  // A row index (token within tile) and B col index
  const int h = lane >> 4;  // K-half selector
  const long tokenBase = (long)wave * 16;

  const bool bvalid = (n < NE);
  const float* ap = x + (tokenBase + n) * ND + 2 * h;
  const float* bp = W + (long)(bvalid ? n : 0) * ND + 2 * h;  // clamped: always valid

  v8f c = {0.f, 0.f, 0.f, 0.f, 0.f, 0.f, 0.f, 0.f};
#pragma unroll 8
  for (int k0 = 0; k0 < ND; k0 += 4) {
    const v2f a = *(const v2f*)(ap + k0);
    const v2f bm = *(const v2f*)(bp + k0);  // unconditional: no EXEC branching
    // D = A(16x4,f32) * B(4x16,f32) + C(16x16,f32)
    c = __builtin_amdgcn_wmma_f32_16x16x4_f32(false, a, false, bm, (short)0, c,
                                              false, false);
  }

  if (bvalid) {
    const float bb = bias[n];
#pragma unroll
    for (int v = 0; v < 8; ++v) {
      const long token = tokenBase + v + 8 * h;
      logits[token * NE + n] = c[v] + bb;
    }
  }
}

// ---------------------------------------------------------------------------
// Kernel 2: one 1024-thread block per batch row.
// Each thread owns tokens [4*tid, 4*tid+4) fully in registers.
// Phase A: softmax over E=4.
// Phase B: for j = 3..0 with caps {2048,1024,512,512}: radix-select the
//   cap-th largest prob among unassigned tokens (keys = raw float bits,
//   assigned -> 0), assign all keys > theta, fill ties at theta in index
//   order via a block-wide exclusive scan (matches top_k tie-breaking).
// ---------------------------------------------------------------------------
__global__ __launch_bounds__(1024) void router_assign(
    const float* __restrict__ logits, int* __restrict__ mask_out,
    float* __restrict__ prob_out) {
  const int b = blockIdx.x;
  const int tid = threadIdx.x;
  const int lane = tid & 31;
  const int wid = tid >> 5;
  const int tok0 = tid * 4;

  __shared__ unsigned hist[256];
  __shared__ unsigned wbase[32];
  __shared__ unsigned sh_prefix, sh_r;

  float pr[4][NE];
  int msk[4];
  bool asg[4];

#pragma unroll
  for (int t = 0; t < 4; ++t) {
    const float* lp = logits + ((long)b * NT + tok0 + t) * NE;
    const float l0 = lp[0], l1 = lp[1], l2 = lp[2], l3 = lp[3];
    const float m = fmaxf(fmaxf(l0, l1), fmaxf(l2, l3));
    const float e0 = expf(l0 - m), e1 = expf(l1 - m);
    const float e2 = expf(l2 - m), e3 = expf(l3 - m);
    const float inv = 1.f / (e0 + e1 + e2 + e3);
    pr[t][0] = e0 * inv;
    pr[t][1] = e1 * inv;
    pr[t][2] = e2 * inv;
    pr[t][3] = e3 * inv;
    msk[t] = 0;
    asg[t] = false;
  }

  const int caps[NE] = {NT / 8, NT / 8, NT / 4, NT / 2};

#pragma unroll
  for (int j = NE - 1; j >= 0; --j) {
    unsigned key[4];
#pragma unroll
    for (int t = 0; t < 4; ++t)
      key[t] = asg[t] ? 0u : __float_as_uint(pr[t][j]);

    // --- 4-pass MSB-first radix select for the caps[j]-th largest key ---
    unsigned prefix = 0;
    unsigned r = (unsigned)caps[j];
    for (int pass = 0; pass < 4; ++pass) {
      const int shift = 24 - pass * 8;
      if (tid < 256) hist[tid] = 0u;
      __syncthreads();
#pragma unroll
      for (int t = 0; t < 4; ++t) {
        const unsigned k = key[t];
        if (pass == 0 || (k >> (shift + 8)) == prefix)
          atomicAdd(&hist[(k >> shift) & 0xFFu], 1u);
      }
      __syncthreads();
      if (tid == 0) {
        unsigned cum = 0;
        int d = 255;
        for (; d >= 0; --d) {
          if (cum + hist[d] >= r) break;
          cum += hist[d];
        }
        if (d < 0) d = 0;  // unreachable: candidates >= r
        r -= cum;          // rank within the ==digit group
        prefix = (prefix << 8) | (unsigned)d;
        sh_prefix = prefix;
        sh_r = r;
      }
      __syncthreads();
      prefix = sh_prefix;
      r = sh_r;
    }
    const unsigned theta = prefix;  // value of the caps[j]-th largest key
    // r = how many of the ==theta keys to take, in index order.

    // --- block-wide exclusive scan of (#tokens == theta) per thread ---
    unsigned cnt = 0;
#pragma unroll
    for (int t = 0; t < 4; ++t) cnt += (key[t] == theta) ? 1u : 0u;
    unsigned inc = cnt;
#pragma unroll
    for (int o = 1; o < 32; o <<= 1) {
      const unsigned v = __shfl_up(inc, o);
      if (lane >= o) inc += v;
    }
    if (lane == 31) wbase[wid] = inc;
    __syncthreads();
    if (wid == 0) {
      const unsigned tot = wbase[lane];
      unsigned inc2 = tot;
#pragma unroll
      for (int o = 1; o < 32; o <<= 1) {
        const unsigned v = __shfl_up(inc2, o);
        if (lane >= o) inc2 += v;
      }
      wbase[lane] = inc2 - tot;  // exclusive base per wave
    }
    __syncthreads();
    unsigned off = wbase[wid] + (inc - cnt);

#pragma unroll
    for (int t = 0; t < 4; ++t) {
      if (asg[t]) continue;
      if (key[t] > theta) {
        msk[t] = j;
        asg[t] = true;
      } else if (key[t] == theta) {
        if (off < r) {
          msk[t] = j;
          asg[t] = true;
        }
        ++off;
      }
    }
    __syncthreads();
  }

#pragma unroll
  for (int t = 0; t < 4; ++t) {
    const long g = (long)b * NT + tok0 + t;
    mask_out[g] = msk[t];
    prob_out[g] = pr[t][msk[t]];
  }
}

// ---------------------------------------------------------------------------
extern "C" void kernel_launch(void* const* d_in, const int* in_sizes, int n_in,
                              void* d_out, int out_size, void* d_ws,
                              size_t ws_size, hipStream_t stream) {
  const float* x = (const float*)d_in[0];     // [NB, NT, ND] f32
  const float* W = (const float*)d_in[1];     // [NE, ND] f32
  const float* bias = (const float*)d_in[2];  // [NE] f32

  float* logits = (float*)d_ws;  // NB*NT*NE f32 = 512 KB scratch

  // d_out: [token_mask int32 (NB*NT)] ++ [expert_probs f32 (NB*NT)]
  int* mask_out = (int*)d_out;
  float* prob_out = (float*)d_out + (long)NB * NT;

  // 2048 waves total, 8 waves (256 threads) per block -> 256 blocks
  router_logits_wmma<<<dim3((NB * NT / 16) / 8), 256, 0, stream>>>(x, W, bias,
                                                                   logits);
  router_assign<<<dim3(NB), 1024, 0, stream>>>(logits, mask_out, prob_out);

  (void)in_sizes;
  (void)n_in;
  (void)out_size;
  (void)ws_size;
}